// GATRatingPrediction_37933151158966
// MI455X (gfx1250) — compile-verified
//
#include <hip/hip_runtime.h>
#include <hip/hip_bf16.h>

typedef float v2f __attribute__((ext_vector_type(2)));
typedef float v8f __attribute__((ext_vector_type(8)));

#define NEG_SLOPE 0.2f

// ---------------------------------------------------------------------------
// Monotone float <-> uint key for atomicMax-based segment max
// ---------------------------------------------------------------------------
__device__ __forceinline__ unsigned fkey(float x) {
  unsigned b = __float_as_uint(x);
  return (b & 0x80000000u) ? ~b : (b | 0x80000000u);
}
__device__ __forceinline__ float funkey(unsigned k) {
  return (k & 0x80000000u) ? __uint_as_float(k & 0x7FFFFFFFu) : __uint_as_float(~k);
}
// key of -inf: fkey(0xFF800000) = ~0xFF800000 = 0x007FFFFF
#define NEG_INF_KEY 0x007FFFFFu

// ---------------------------------------------------------------------------
// Fill a u32 buffer (used for zeroing accumulators and seeding emax)
// ---------------------------------------------------------------------------
__global__ void fill_u32_kernel(unsigned* __restrict__ p, unsigned v, int n) {
  int i = blockIdx.x * blockDim.x + threadIdx.x;
  if (i < n) p[i] = v;
}

// ---------------------------------------------------------------------------
// fp32 GEMM on the matrix core: C[M,Ncol] = A[M,K] @ B[K,Ncol]
// One wave per 16x16 output tile, V_WMMA_F32_16X16X4_F32, K stepped by 4.
// Requires M % 16 == 0, Ncol % 16 == 0, blockDim.x == 32*(Ncol/16).
// A-frag layout: lanes 0-15 -> K=k0,k0+1 ; lanes 16-31 -> K=k0+2,k0+3.
// B-frag layout: mirrored (row of B striped across lanes).
// D layout: VGPR r -> M = r (lanes 0-15), M = 8+r (lanes 16-31).
// ---------------------------------------------------------------------------
__global__ void gemm_wmma_f32_kernel(const float* __restrict__ A,
                                     const float* __restrict__ B,
                                     float* __restrict__ C,
                                     int M, int K, int Ncol) {
  const int wave = threadIdx.x >> 5;
  const int lane = threadIdx.x & 31;
  const int rowBase = blockIdx.x * 16;
  const int colBase = wave * 16;
  const int lm = lane & 15;
  const int hi = lane >> 4;   // 0 or 1 (upper half-wave handles K+2,K+3)

  v8f acc = {};
  const float* arow = A + (size_t)(rowBase + lm) * K;
  for (int k0 = 0; k0 < K; k0 += 4) {
    const int ka = k0 + hi * 2;
    v2f a, b;
    a.x = arow[ka];
    a.y = arow[ka + 1];
    b.x = B[(size_t)ka * Ncol + colBase + lm];
    b.y = B[(size_t)(ka + 1) * Ncol + colBase + lm];
    acc = __builtin_amdgcn_wmma_f32_16x16x4_f32(
        /*neg_a=*/false, a, /*neg_b=*/false, b,
        /*c_mod=*/(short)0, acc, /*reuse_a=*/false, /*reuse_b=*/false);
  }
#pragma unroll
  for (int r = 0; r < 8; ++r) {
    const int row = rowBase + hi * 8 + r;
    C[(size_t)row * Ncol + colBase + lm] = acc[r];
  }
}

// ---------------------------------------------------------------------------
// alpha_src[n,h] = <h[n,h,:], a_src[h,:]>  (same for a_dst)
// ---------------------------------------------------------------------------
__global__ void alpha_kernel(const float* __restrict__ h,
                             const float* __restrict__ a_src,
                             const float* __restrict__ a_dst,
                             float* __restrict__ as, float* __restrict__ ad,
                             int N, int H, int D) {
  int i = blockIdx.x * blockDim.x + threadIdx.x;
  if (i >= N * H) return;
  int hh = i % H;
  const float* hp = h + (size_t)i * D;
  const float* sp = a_src + (size_t)hh * D;
  const float* dp = a_dst + (size_t)hh * D;
  float s = 0.f, d = 0.f;
  for (int k = 0; k < D; ++k) { float v = hp[k]; s += v * sp[k]; d += v * dp[k]; }
  as[i] = s; ad[i] = d;
}

__device__ __forceinline__ void edge_sd(const int* __restrict__ ei, int E, int i,
                                        int& s, int& d) {
  if (i < E) { s = ei[i]; d = ei[E + i]; } else { s = d = i - E; }  // self loops
}

// ---------------------------------------------------------------------------
// Pass 1: segment max of leaky_relu(as[src]+ad[dst]) over incoming edges
// ---------------------------------------------------------------------------
__global__ void edge_max_kernel(const int* __restrict__ ei, int E, int N,
                                const float* __restrict__ as,
                                const float* __restrict__ ad,
                                unsigned* __restrict__ emax, int H) {
  int i = blockIdx.x * blockDim.x + threadIdx.x;
  if (i >= E + N) return;
  int s, d; edge_sd(ei, E, i, s, d);
  for (int h = 0; h < H; ++h) {
    float e = as[s * H + h] + ad[d * H + h];
    e = (e >= 0.f) ? e : NEG_SLOPE * e;
    atomicMax(&emax[d * H + h], fkey(e));
  }
}

// ---------------------------------------------------------------------------
// Pass 2: ex = exp(e - max); esum[dst,h] += ex; acc[dst,h,:] += ex * h[src,h,:]
// One thread per (edge, head).
// ---------------------------------------------------------------------------
__global__ void edge_acc_kernel(const int* __restrict__ ei, int E, int N,
                                const float* __restrict__ as,
                                const float* __restrict__ ad,
                                const unsigned* __restrict__ emax,
                                float* __restrict__ esum,
                                const float* __restrict__ hfeat,
                                float* __restrict__ acc,
                                int H, int D) {
  int t = blockIdx.x * blockDim.x + threadIdx.x;
  if (t >= (E + N) * H) return;
  int i = t / H, h = t % H;
  int s, d; edge_sd(ei, E, i, s, d);
  float e = as[s * H + h] + ad[d * H + h];
  e = (e >= 0.f) ? e : NEG_SLOPE * e;
  float ex = expf(e - funkey(emax[d * H + h]));
  atomicAdd(&esum[d * H + h], ex);
  const float* hs = hfeat + ((size_t)s * H + h) * D;
  float* ap = acc + ((size_t)d * H + h) * D;
  for (int k = 0; k < D; ++k) atomicAdd(&ap[k], ex * hs[k]);
}

// ---------------------------------------------------------------------------
// Finalize: out = acc/(esum+1e-16) + bias, optional ELU. In-place on acc.
// ---------------------------------------------------------------------------
__global__ void finalize_kernel(float* __restrict__ acc,
                                const float* __restrict__ esum,
                                const float* __restrict__ bias,
                                int N, int H, int D, int apply_elu) {
  int i = blockIdx.x * blockDim.x + threadIdx.x;
  int HD = H * D;
  if (i >= N * HD) return;
  int n = i / HD, c = i % HD, h = c / D;
  float v = acc[i] / (esum[n * H + h] + 1e-16f) + bias[c];
  if (apply_elu) v = (v > 0.f) ? v : (expf(v) - 1.f);
  acc[i] = v;
}

// ---------------------------------------------------------------------------
// Rating head, factored: ps[n] = <h2[n], Wp[:D]>, pd[n] = <h2[n], Wp[D:]>
// ---------------------------------------------------------------------------
__global__ void pred_node_kernel(const float* __restrict__ h2,
                                 const float* __restrict__ Wp,
                                 float* __restrict__ ps, float* __restrict__ pd,
                                 int N, int D) {
  int n = blockIdx.x * blockDim.x + threadIdx.x;
  if (n >= N) return;
  const float* hp = h2 + (size_t)n * D;
  float s = 0.f, d = 0.f;
  for (int k = 0; k < D; ++k) { s += hp[k] * Wp[k]; d += hp[k] * Wp[D + k]; }
  ps[n] = s; pd[n] = d;
}

__global__ void pred_edge_kernel(const int* __restrict__ ei, int E,
                                 const float* __restrict__ ps,
                                 const float* __restrict__ pd,
                                 const float* __restrict__ bp,
                                 float* __restrict__ out) {
  int e = blockIdx.x * blockDim.x + threadIdx.x;
  if (e >= E) return;
  out[e] = ps[ei[e]] + pd[ei[E + e]] + bp[0];
}

// ---------------------------------------------------------------------------
// Host-side launch
// ---------------------------------------------------------------------------
static inline int cdiv(long long a, int b) { return (int)((a + b - 1) / b); }

extern "C" void kernel_launch(void* const* d_in, const int* in_sizes, int n_in,
                              void* d_out, int out_size, void* d_ws, size_t ws_size,
                              hipStream_t stream) {
  const float* x   = (const float*)d_in[0];
  const int*   ei  = (const int*)  d_in[1];
  const float* W1  = (const float*)d_in[2];
  const float* as1w= (const float*)d_in[3];
  const float* ad1w= (const float*)d_in[4];
  const float* b1  = (const float*)d_in[5];
  const float* W2  = (const float*)d_in[6];
  const float* as2w= (const float*)d_in[7];
  const float* ad2w= (const float*)d_in[8];
  const float* b2  = (const float*)d_in[9];
  const float* Wp  = (const float*)d_in[10];
  const float* bp  = (const float*)d_in[11];
  float* out = (float*)d_out;

  const int F = 128, H1 = 4, Hd = 32;           // fixed by the reference
  const int N = in_sizes[0] / F;                // 50000
  const int E = in_sizes[1] / 2;                // 800000
  const int EN = E + N;                         // edges incl. self loops

  // ---- workspace carve-up (floats) ----
  float* ws = (float*)d_ws;
  float*    h1    = ws;                 ws += (size_t)N * F;        // [N,128]
  float*    acc1  = ws;                 ws += (size_t)N * F;        // [N,128] -> layer1 out
  float*    as1   = ws;                 ws += (size_t)N * H1;
  float*    ad1   = ws;                 ws += (size_t)N * H1;
  unsigned* emax1 = (unsigned*)ws;      ws += (size_t)N * H1;
  float*    esum1 = ws;                 ws += (size_t)N * H1;
  float*    h2p   = ws;                 ws += (size_t)N * Hd;       // [N,32]
  float*    acc2  = ws;                 ws += (size_t)N * Hd;       // [N,32] -> layer2 out
  float*    as2   = ws;                 ws += (size_t)N;
  float*    ad2   = ws;                 ws += (size_t)N;
  unsigned* emax2 = (unsigned*)ws;      ws += (size_t)N;
  float*    esum2 = ws;                 ws += (size_t)N;
  float*    ps    = ws;                 ws += (size_t)N;
  float*    pd    = ws;                 ws += (size_t)N;
  (void)ws_size;

  const int T = 256;

  // ---- init accumulators (deterministic per call) ----
  fill_u32_kernel<<<cdiv((long long)N * F, T), T, 0, stream>>>((unsigned*)acc1, 0u, N * F);
  fill_u32_kernel<<<cdiv((long long)N * H1, T), T, 0, stream>>>((unsigned*)esum1, 0u, N * H1);
  fill_u32_kernel<<<cdiv((long long)N * H1, T), T, 0, stream>>>(emax1, NEG_INF_KEY, N * H1);
  fill_u32_kernel<<<cdiv((long long)N * Hd, T), T, 0, stream>>>((unsigned*)acc2, 0u, N * Hd);
  fill_u32_kernel<<<cdiv(N, T), T, 0, stream>>>((unsigned*)esum2, 0u, N);
  fill_u32_kernel<<<cdiv(N, T), T, 0, stream>>>(emax2, NEG_INF_KEY, N);

  // ---- layer 1 ----
  // h1 = x @ W1 : M=N (divisible by 16), K=128, Ncol=128 -> 8 waves/block
  gemm_wmma_f32_kernel<<<N / 16, 8 * 32, 0, stream>>>(x, W1, h1, N, F, F);
  alpha_kernel<<<cdiv((long long)N * H1, T), T, 0, stream>>>(h1, as1w, ad1w, as1, ad1, N, H1, Hd);
  edge_max_kernel<<<cdiv(EN, T), T, 0, stream>>>(ei, E, N, as1, ad1, emax1, H1);
  edge_acc_kernel<<<cdiv((long long)EN * H1, T), T, 0, stream>>>(ei, E, N, as1, ad1, emax1,
                                                                 esum1, h1, acc1, H1, Hd);
  finalize_kernel<<<cdiv((long long)N * F, T), T, 0, stream>>>(acc1, esum1, b1, N, H1, Hd, 1);

  // ---- layer 2 ----
  // h2p = acc1 @ W2 : K=128, Ncol=32 -> 2 waves/block
  gemm_wmma_f32_kernel<<<N / 16, 2 * 32, 0, stream>>>(acc1, W2, h2p, N, F, Hd);
  alpha_kernel<<<cdiv(N, T), T, 0, stream>>>(h2p, as2w, ad2w, as2, ad2, N, 1, Hd);
  edge_max_kernel<<<cdiv(EN, T), T, 0, stream>>>(ei, E, N, as2, ad2, emax2, 1);
  edge_acc_kernel<<<cdiv(EN, T), T, 0, stream>>>(ei, E, N, as2, ad2, emax2,
                                                 esum2, h2p, acc2, 1, Hd);
  finalize_kernel<<<cdiv((long long)N * Hd, T), T, 0, stream>>>(acc2, esum2, b2, N, 1, Hd, 0);

  // ---- rating head ----
  pred_node_kernel<<<cdiv(N, T), T, 0, stream>>>(acc2, Wp, ps, pd, N, Hd);
  pred_edge_kernel<<<cdiv(E, T), T, 0, stream>>>(ei, E, ps, pd, bp, out);
}